// SolveWeightLST2d_88510686036693
// MI455X (gfx1250) — compile-verified
//
#include <hip/hip_runtime.h>
#include <hip/hip_bf16.h>

typedef __attribute__((ext_vector_type(2))) float v2f;
typedef __attribute__((ext_vector_type(8))) float v8f;

#define RIDGE_F 1e-7f
#define WLIM_F  5000.0f

__device__ __forceinline__ void atomic_add_f32(float* p, float v) {
#if __has_builtin(__builtin_amdgcn_global_atomic_fadd_f32)
    __builtin_amdgcn_global_atomic_fadd_f32(p, v);
#else
    atomicAdd(p, v);
#endif
}

// ---------------------------------------------------------------------------
// Phase 0: zero the per-node upper-triangular accumulators (N*15 floats).
// ---------------------------------------------------------------------------
__global__ void swl_zero_kernel(float* __restrict__ p, long long n) {
    long long i = (long long)blockIdx.x * blockDim.x + threadIdx.x;
    long long stride = (long long)gridDim.x * blockDim.x;
    for (; i < n; i += stride) p[i] = 0.0f;
}

// ---------------------------------------------------------------------------
// Phase 1: per-edge outer-product scatter.  M_node[row] += H H^T  (upper tri,
// 15 unique entries; symmetric => mirror at solve time).  48M f32 atomics into
// a 6MB region => fully L2-resident non-returning atomic adds.
// ---------------------------------------------------------------------------
__global__ __launch_bounds__(256) void swl_accum_kernel(
        const float* __restrict__ pos,
        const long long* __restrict__ eidx,
        float* __restrict__ M, long long E) {
    long long e = (long long)blockIdx.x * blockDim.x + threadIdx.x;
    if (e >= E) return;
    long long r = __builtin_nontemporal_load(&eidx[e]);
    long long c = __builtin_nontemporal_load(&eidx[E + e]);
    const float2* p2 = (const float2*)pos;
    float2 pr = p2[r];
    float2 pc = p2[c];
    float x = pc.x - pr.x;
    float y = pc.y - pr.y;
    float H[5] = { x, y, x * y, x * x, y * y };
    float* base = M + r * 15;
#pragma unroll
    for (int j = 0; j < 5; ++j) {
#pragma unroll
        for (int i = 0; i <= j; ++i) {
            atomic_add_f32(&base[j * (j + 1) / 2 + i], H[i] * H[j]);
        }
    }
}

// ---------------------------------------------------------------------------
// Phase 2: per-node 5x5 SPD solve via fully unrolled Cholesky.
// M is SPD (normal-equation sum + ridge*I), so no pivoting needed.
// rhs = [0,0,0,2,2].  Writes C (N x 5).
// ---------------------------------------------------------------------------
__global__ __launch_bounds__(256) void swl_solve_kernel(
        const float* __restrict__ M,
        float* __restrict__ C, int N) {
    int n = blockIdx.x * blockDim.x + threadIdx.x;
    if (n >= N) return;
    const float* t = M + (long long)n * 15;
    float A[5][5];
#pragma unroll
    for (int j = 0; j < 5; ++j) {
#pragma unroll
        for (int i = 0; i <= j; ++i) {
            float v = t[j * (j + 1) / 2 + i];
            A[i][j] = v;
            A[j][i] = v;
        }
    }
#pragma unroll
    for (int d = 0; d < 5; ++d) A[d][d] += RIDGE_F;

    // Cholesky: A = L L^T, L stored in lower triangle.
#pragma unroll
    for (int k = 0; k < 5; ++k) {
        float s = sqrtf(A[k][k]);
        float inv = 1.0f / s;
        A[k][k] = s;
#pragma unroll
        for (int i = k + 1; i < 5; ++i) A[i][k] *= inv;
#pragma unroll
        for (int j = k + 1; j < 5; ++j) {
#pragma unroll
            for (int i = j; i < 5; ++i) A[i][j] -= A[i][k] * A[j][k];
        }
    }
    const float b[5] = { 0.0f, 0.0f, 0.0f, 2.0f, 2.0f };
    float z[5];
#pragma unroll
    for (int i = 0; i < 5; ++i) {
        float s = b[i];
#pragma unroll
        for (int j = 0; j < 5; ++j) if (j < i) s -= A[i][j] * z[j];
        z[i] = s / A[i][i];
    }
    float cvec[5];
#pragma unroll
    for (int i = 4; i >= 0; --i) {
        float s = z[i];
#pragma unroll
        for (int j = 0; j < 5; ++j) if (j > i) s -= A[j][i] * cvec[j];
        cvec[i] = s / A[i][i];
    }
#pragma unroll
    for (int i = 0; i < 5; ++i) C[(long long)n * 5 + i] = cvec[i];
}

// ---------------------------------------------------------------------------
// Phase 3: per-edge weights via V_WMMA_F32_16X16X4_F32 (exact f32 MACs).
// Each wave handles 16 edges.  K = basis components 0..3:
//   A[m,k] = H_{e_m}[k]  (16x4: lanes 0-15 carry K=0,1; lanes 16-31 K=2,3;
//                         lane m and m+16 just recompute the same edge's H)
//   B[k,n] = C[row_{e_n}][k]  (4x16, packed the same way)
//   D[n,n] = sum_{k<4} H_{e_n}[k] * C[row_{e_n}][k]
// 5th component (y^2 * c4) added scalar, then clamp and store.
// Diagonal D[n,n] lives at lane n (n<8, VGPR n) / lane n+16 (n>=8, VGPR n-8),
// i.e. VGPR (lane&7) on lanes 0-7 and 24-31.
// ---------------------------------------------------------------------------
__global__ __launch_bounds__(256) void swl_weights_kernel(
        const float* __restrict__ pos,
        const long long* __restrict__ eidx,
        const float* __restrict__ C,
        float* __restrict__ out, long long E) {
    const int lane = threadIdx.x & 31;
    const int wave = threadIdx.x >> 5;
    const long long tile = ((long long)blockIdx.x * (blockDim.x >> 5) + wave) * 16;
    const int slot = lane & 15;
    long long ge = tile + slot;
    long long e = (ge < E) ? ge : (E - 1);   // clamp so EXEC stays all-ones

    long long r = __builtin_nontemporal_load(&eidx[e]);
    long long c = __builtin_nontemporal_load(&eidx[E + e]);
    const float2* p2 = (const float2*)pos;
    float2 pr = p2[r];
    float2 pc = p2[c];
    float x = pc.x - pr.x;
    float y = pc.y - pr.y;
    float H0 = x, H1 = y, H2 = x * y, H3 = x * x, H4 = y * y;

    const float* Cr = C + r * 5;
    float c0 = Cr[0], c1 = Cr[1], c2 = Cr[2], c3 = Cr[3], c4 = Cr[4];

    const bool hi = lane >= 16;
    v2f a, bm;
    a.x  = hi ? H2 : H0;   // A[m,2] : A[m,0]
    a.y  = hi ? H3 : H1;   // A[m,3] : A[m,1]
    bm.x = hi ? c2 : c0;   // B[2,n] : B[0,n]
    bm.y = hi ? c3 : c1;   // B[3,n] : B[1,n]

    v8f acc = {};
    v8f d = __builtin_amdgcn_wmma_f32_16x16x4_f32(
        /*neg_a=*/false, a, /*neg_b=*/false, bm,
        /*c_mod=*/(short)0, acc, /*reuse_a=*/false, /*reuse_b=*/false);

    // Extract D[n,n] from VGPR (lane & 7); only lanes 0-7 and 24-31 hold it.
    int k = lane & 7;
    float w = d[0];
    w = (k == 1) ? d[1] : w;
    w = (k == 2) ? d[2] : w;
    w = (k == 3) ? d[3] : w;
    w = (k == 4) ? d[4] : w;
    w = (k == 5) ? d[5] : w;
    w = (k == 6) ? d[6] : w;
    w = (k == 7) ? d[7] : w;

    w += H4 * c4;                           // 5th basis component
    w = fminf(fmaxf(w, -WLIM_F), WLIM_F);   // clip

    bool active = (lane < 8) || (lane >= 24);
    if (active && ge < E) __builtin_nontemporal_store(w, &out[ge]);
}

// ---------------------------------------------------------------------------
extern "C" void kernel_launch(void* const* d_in, const int* in_sizes, int n_in,
                              void* d_out, int out_size, void* d_ws, size_t ws_size,
                              hipStream_t stream) {
    const float* pos = (const float*)d_in[0];
    const long long* eidx = (const long long*)d_in[1];
    const int N = in_sizes[0] / 2;            // 100000 nodes
    const long long E = (long long)in_sizes[1] / 2;  // 3.2M edges
    float* out = (float*)d_out;

    // Workspace layout: M (N*15 f32) | C (N*5 f32)
    float* M = (float*)d_ws;
    size_t m_bytes = ((size_t)N * 15 * sizeof(float) + 255) & ~(size_t)255;
    float* C = (float*)((char*)d_ws + m_bytes);

    // Phase 0: zero accumulators
    long long mcount = (long long)N * 15;
    int zblocks = (int)((mcount + 255) / 256);
    hipLaunchKernelGGL(swl_zero_kernel, dim3(zblocks), dim3(256), 0, stream, M, mcount);

    // Phase 1: scatter HH^T
    int ablocks = (int)((E + 255) / 256);
    hipLaunchKernelGGL(swl_accum_kernel, dim3(ablocks), dim3(256), 0, stream,
                       pos, eidx, M, E);

    // Phase 2: per-node Cholesky solve
    int sblocks = (N + 255) / 256;
    hipLaunchKernelGGL(swl_solve_kernel, dim3(sblocks), dim3(256), 0, stream,
                       M, C, N);

    // Phase 3: per-edge weights via f32 WMMA (16 edges per wave, 8 waves/block)
    long long waves = (E + 15) / 16;
    int wblocks = (int)((waves + 7) / 8);
    hipLaunchKernelGGL(swl_weights_kernel, dim3(wblocks), dim3(256), 0, stream,
                       pos, eidx, C, out, E);
}